// NMMultivariateGaussianMixture_2662879723705
// MI455X (gfx1250) — compile-verified
//
#include <hip/hip_runtime.h>
#include <math.h>

typedef __attribute__((ext_vector_type(2))) float v2f;
typedef __attribute__((ext_vector_type(8))) float v8f;

#define NK 8
#define ND 32
#define NP 64              // NK*NK cluster pairs
#define LOG2PI 1.8378770664093453f

// workspace layout (in floats)
#define OFF_T    0                       // T[64][32][32]  = inv(sigma_p) = invLc_i + invLc_j
#define OFF_B    (OFF_T + NP*ND*ND)      // b[64][32]      = T_p @ mu_p  (== rhs)
#define OFF_C    (OFF_B + NP*ND)         // c[64]          per-pair constant
#define OFF_INV  (OFF_C + NP)            // invLc[8][32][32] scratch
#define OFF_LL   (OFF_INV + NK*ND*ND)    // ll[N] per-sample log-likelihood

// ---------------------------------------------------------------------------
// Kernel 1: tiny setup. A_k = L_k L_k^T + I ; Lc = chol(A) ; invLc = inv(Lc);
// per pair: T = invLc_i+invLc_j (lower-tri), b = invLc_i mu_i + invLc_j mu_j,
// c = z + logw_i + logw_j - 0.5*D*LOG2PI + 0.5*log det(T)
// ---------------------------------------------------------------------------
__global__ __launch_bounds__(256) void setup_kernel(const float* __restrict__ mu,
                                                    const float* __restrict__ L,
                                                    const float* __restrict__ w,
                                                    float* __restrict__ ws) {
  __shared__ float Lc[NK][ND][ND];   // 32 KB
  __shared__ float logw[NK];
  const int tid = threadIdx.x;

  // Phase A: A_k = L_k L_k^T + I  (8192 entries over 256 threads)
  for (int idx = tid; idx < NK * ND * ND; idx += 256) {
    int k = idx >> 10, r = (idx >> 5) & 31, c = idx & 31;
    const float* Lk = L + k * ND * ND;
    float s = (r == c) ? 1.0f : 0.0f;
    for (int e = 0; e < ND; ++e) s += Lk[r * ND + e] * Lk[c * ND + e];
    Lc[k][r][c] = s;
  }
  __syncthreads();

  // Phase B: in-place Cholesky (lower), 8 groups x 32 threads, lockstep in j
  {
    const int k = tid >> 5, tx = tid & 31;
    for (int j = 0; j < ND; ++j) {
      if (tx == j) {
        float d = Lc[k][j][j];
        for (int p = 0; p < j; ++p) d -= Lc[k][j][p] * Lc[k][j][p];
        Lc[k][j][j] = sqrtf(d);
      }
      __syncthreads();
      if (tx > j) {
        float s = Lc[k][tx][j];
        for (int p = 0; p < j; ++p) s -= Lc[k][tx][p] * Lc[k][j][p];
        Lc[k][tx][j] = s / Lc[k][j][j];
      }
      __syncthreads();
    }
  }

  // Phase C: invLc by forward substitution, one column per thread (8x32 = 256)
  {
    const int k = tid >> 5, j = tid & 31;
    float* inv = ws + OFF_INV + k * ND * ND;
    float col[ND];
    for (int i = 0; i < ND; ++i) col[i] = 0.0f;
    col[j] = 1.0f / Lc[k][j][j];
    for (int i = j + 1; i < ND; ++i) {
      float s = 0.0f;
      for (int p = j; p < i; ++p) s -= Lc[k][i][p] * col[p];
      col[i] = s / Lc[k][i][i];
    }
    for (int i = 0; i < ND; ++i) inv[i * ND + j] = col[i];
  }

  // log-softmax of weights
  if (tid == 0) {
    float mx = -INFINITY;
    for (int k = 0; k < NK; ++k) mx = fmaxf(mx, w[k]);
    float s = 0.0f;
    for (int k = 0; k < NK; ++k) s += expf(w[k] - mx);
    float lse = mx + logf(s);
    for (int k = 0; k < NK; ++k) logw[k] = w[k] - lse;
  }
  __syncthreads();

  // Phase D: T[p] = invLc_i + invLc_j  (65536 entries over 256 threads)
  const float* invG = ws + OFF_INV;
  for (int idx = tid; idx < NP * ND * ND; idx += 256) {
    int p = idx >> 10, rem = idx & 1023;
    int pi = p >> 3, pj = p & 7;
    ws[OFF_T + idx] = invG[pi * 1024 + rem] + invG[pj * 1024 + rem];
  }

  // Phase E: per-pair b and constant c (64 threads)
  if (tid < NP) {
    const int p = tid, i = p >> 3, j = p & 7;
    const float* invi = invG + i * 1024;
    const float* invj = invG + j * 1024;
    const float* mui = mu + i * ND;
    const float* muj = mu + j * ND;

    float* bp = ws + OFF_B + p * ND;
    for (int d = 0; d < ND; ++d) {
      float s = 0.0f;
      for (int e = 0; e <= d; ++e)
        s += invi[d * ND + e] * mui[e] + invj[d * ND + e] * muj[e];
      bp[d] = s;
    }

    // forward solve (Lc_i + Lc_j) y = dmu ; accumulate dets
    float y[ND];
    float logdetS = 0.0f, logdetT = 0.0f, quad = 0.0f;
    for (int d = 0; d < ND; ++d) {
      float dm = mui[d] - muj[d];
      float s = dm;
      for (int e = 0; e < d; ++e) s -= (Lc[i][d][e] + Lc[j][d][e]) * y[e];
      float Sdd = Lc[i][d][d] + Lc[j][d][d];
      y[d] = s / Sdd;
      quad += dm * y[d];
      logdetS += logf(Sdd);
      logdetT += logf(invi[d * ND + d] + invj[d * ND + d]);
    }
    quad *= -0.5f;
    float z = -0.5f * ((float)ND * LOG2PI + logdetS) + quad;
    ws[OFF_C + p] = z + logw[i] + logw[j]
                    - 0.5f * (float)ND * LOG2PI + 0.5f * logdetT;
  }
}

// ---------------------------------------------------------------------------
// Kernel 2: the heavy GEMM, fused with the per-sample streaming logsumexp.
// m[32 x 16] = T_p[32x32] @ X_tile^T[32x16] via V_WMMA_F32_16X16X4_F32.
// Each wave owns 4 sample-tiles (64 samples) so each T_p fragment load is
// amortized over 64 WMMAs. 8 waves / block -> 512 samples per block.
// ---------------------------------------------------------------------------
#define STILE 4
__global__ __launch_bounds__(256) void gmm_main_kernel(const float* __restrict__ X,
                                                       const float* __restrict__ ws,
                                                       float* __restrict__ ll,
                                                       int N) {
  const float* __restrict__ T  = ws + OFF_T;
  const float* __restrict__ Bv = ws + OFF_B;
  const float* __restrict__ Cv = ws + OFF_C;

  const int lane = threadIdx.x & 31;
  const int wave = threadIdx.x >> 5;
  const int r = lane & 15;              // column (sample) / row-within-tile
  const int hi = lane >> 4;             // half-wave select
  const int hioff = hi << 1;            // K offset: 0 or 2 per ISA A/B striping
  const int hi8 = hi << 3;              // C/D row offset: 0 or 8
  const int s0 = (blockIdx.x * 8 + wave) * (16 * STILE);

  // B fragments: Xt[K, sample]; per ISA lane r holds column r,
  // vgpr0 = K(+0/+2 by half-wave), vgpr1 = K(+1/+3).  float2 global loads.
  v2f xb[STILE][8];
#pragma unroll
  for (int t = 0; t < STILE; ++t) {
    int s = s0 + t * 16 + r;
    if (s >= N) s = N - 1;   // clamp (N is a multiple of 512 in practice)
#pragma unroll
    for (int kk = 0; kk < 8; ++kk)
      xb[t][kk] = *(const v2f*)(X + s * ND + 4 * kk + hioff);
  }

  float m_run[STILE], s_run[STILE];
#pragma unroll
  for (int t = 0; t < STILE; ++t) { m_run[t] = -INFINITY; s_run[t] = 0.0f; }

  for (int p = 0; p < NP; ++p) {
    const float* Tp = T + p * 1024;
    // A fragments: rows 0-15 and 16-31, K blocks of 4
    v2f a0[8], a1[8];
#pragma unroll
    for (int kk = 0; kk < 8; ++kk) {
      a0[kk] = *(const v2f*)(Tp + r * ND + 4 * kk + hioff);
      a1[kk] = *(const v2f*)(Tp + (16 + r) * ND + 4 * kk + hioff);
    }
    const float* bp = Bv + p * ND;
    float b0[8], b1[8];
#pragma unroll
    for (int v = 0; v < 8; ++v) { b0[v] = bp[v + hi8]; b1[v] = bp[16 + v + hi8]; }
    const float cp = Cv[p];

#pragma unroll
    for (int t = 0; t < STILE; ++t) {
      v8f acc0 = {0.f, 0.f, 0.f, 0.f, 0.f, 0.f, 0.f, 0.f};
      v8f acc1 = {0.f, 0.f, 0.f, 0.f, 0.f, 0.f, 0.f, 0.f};
#pragma unroll
      for (int kk = 0; kk < 8; ++kk) {
        acc0 = __builtin_amdgcn_wmma_f32_16x16x4_f32(false, a0[kk], false, xb[t][kk],
                                                     (short)0, acc0, false, false);
        acc1 = __builtin_amdgcn_wmma_f32_16x16x4_f32(false, a1[kk], false, xb[t][kk],
                                                     (short)0, acc1, false, false);
      }
      // ||m - b||^2 : lane holds rows {v+8*hi} and {16+v+8*hi} of column r
      float nh = 0.0f;
#pragma unroll
      for (int v = 0; v < 8; ++v) {
        float d0 = acc0[v] - b0[v];
        float d1 = acc1[v] - b1[v];
        nh += d0 * d0 + d1 * d1;
      }
      float nrm = nh + __shfl_xor(nh, 16);        // join the two half-wave row sets
      float val = cp - 0.5f * nrm;
      // streaming logsumexp over pairs
      float nm = fmaxf(m_run[t], val);
      s_run[t] = s_run[t] * expf(m_run[t] - nm) + expf(val - nm);
      m_run[t] = nm;
    }
  }

#pragma unroll
  for (int t = 0; t < STILE; ++t) {
    int s = s0 + t * 16 + r;
    if (hi == 0 && s < N) ll[s] = m_run[t] + logf(s_run[t]);
  }
}

// ---------------------------------------------------------------------------
// Kernel 3: deterministic single-block logsumexp over ll[N] -> scalar
// ---------------------------------------------------------------------------
__global__ __launch_bounds__(256) void reduce_kernel(const float* __restrict__ ll,
                                                     float* __restrict__ out, int n) {
  __shared__ float sm[256];
  const int tid = threadIdx.x;
  float mx = -INFINITY;
  for (int i = tid; i < n; i += 256) mx = fmaxf(mx, ll[i]);
  sm[tid] = mx;
  __syncthreads();
  for (int s = 128; s > 0; s >>= 1) {
    if (tid < s) sm[tid] = fmaxf(sm[tid], sm[tid + s]);
    __syncthreads();
  }
  const float gmax = sm[0];
  __syncthreads();
  float acc = 0.0f;
  for (int i = tid; i < n; i += 256) acc += expf(ll[i] - gmax);
  __syncthreads();
  sm[tid] = acc;
  __syncthreads();
  for (int s = 128; s > 0; s >>= 1) {
    if (tid < s) sm[tid] += sm[tid + s];
    __syncthreads();
  }
  if (tid == 0) out[0] = gmax + logf(sm[0]);
}

extern "C" void kernel_launch(void* const* d_in, const int* in_sizes, int n_in,
                              void* d_out, int out_size, void* d_ws, size_t ws_size,
                              hipStream_t stream) {
  const float* X  = (const float*)d_in[0];   // [N,32]
  const float* mu = (const float*)d_in[1];   // [8,32]
  const float* L  = (const float*)d_in[2];   // [8,32,32]
  const float* w  = (const float*)d_in[3];   // [8]
  // d_in[4] = it (unused)

  const int N = in_sizes[0] / ND;
  if (N <= 0) return;
  const size_t need = (size_t)(OFF_LL + N) * sizeof(float);
  if (ws_size < need) return;
  float* ws = (float*)d_ws;

  setup_kernel<<<1, 256, 0, stream>>>(mu, L, w, ws);

  const int samples_per_block = 8 * 16 * STILE;   // 512
  const int blocks = (N + samples_per_block - 1) / samples_per_block;
  gmm_main_kernel<<<blocks, 256, 0, stream>>>(X, ws, ws + OFF_LL, N);

  reduce_kernel<<<1, 256, 0, stream>>>(ws + OFF_LL, (float*)d_out, N);
}